// GmapAD_GCN_6700148982129
// MI455X (gfx1250) — compile-verified
//
#include <hip/hip_runtime.h>

#define N_NODES 50000
#define E_EDGES 800000
#define IN_DIM  512
#define HID     256
#define HID2    128

typedef __attribute__((ext_vector_type(2))) float v2f;
typedef __attribute__((ext_vector_type(8))) float v8f;
typedef __attribute__((ext_vector_type(4))) float f4;

#if defined(__gfx1250__) && __has_builtin(__builtin_amdgcn_global_load_async_to_lds_b128)
#define HAVE_ASYNC_COPY 1
typedef int v4i_ __attribute__((vector_size(16)));
typedef __attribute__((address_space(1))) v4i_* gvec_p;   // global int4*
typedef __attribute__((address_space(3))) v4i_* lvec_p;   // LDS int4*
#endif

// ---------------- norm precompute ----------------

__global__ void k_fill1(float* deg) {
    int i = blockIdx.x * 256 + threadIdx.x;
    if (i < N_NODES) deg[i] = 1.0f;          // self-loop contributes 1 to in-degree
}

__global__ void k_degree(const long long* __restrict__ col, float* deg) {
    int e = blockIdx.x * 256 + threadIdx.x;
    if (e < E_EDGES) atomicAdd(&deg[(int)col[e]], 1.0f);
}

__global__ void k_dinv(float* deg) {
    int i = blockIdx.x * 256 + threadIdx.x;
    if (i < N_NODES) deg[i] = rsqrtf(deg[i]);   // deg >= 1 always
}

__global__ void k_norm(const long long* __restrict__ row,
                       const long long* __restrict__ col,
                       const float* __restrict__ dinv, float* norm) {
    int e = blockIdx.x * 256 + threadIdx.x;
    if (e < E_EDGES) norm[e] = dinv[(int)row[e]] * dinv[(int)col[e]];
}

// ---------------- WMMA f32 GEMM: H(M x OUT) = X(M x K) @ W(K x OUT) ----------------
// block = 256 threads = 8 waves; each wave owns one 16x16 output tile.
// gridDim.x = M/16 (M multiple of 16), gridDim.y * 128 covers OUT.

template <int K, int OUT>
__global__ __launch_bounds__(256) void k_gemm_wmma(const float* __restrict__ X,
                                                   const float* __restrict__ W,
                                                   float* __restrict__ H) {
    constexpr int KP = K + 4;                 // pad rows to dodge LDS bank conflicts
    __shared__ float lds_A[16 * KP];

    const int t  = threadIdx.x;
    const int m0 = blockIdx.x * 16;

    // stage A tile (16 rows x K) into LDS, 16B granules
    constexpr int NV = 16 * (K / 4);
    for (int idx = t; idx < NV; idx += 256) {
        int r  = idx / (K / 4);
        int kk = idx % (K / 4);
        const float* gp = X + (size_t)(m0 + r) * K + kk * 4;
        float* lp = &lds_A[r * KP + kk * 4];
#ifdef HAVE_ASYNC_COPY
        __builtin_amdgcn_global_load_async_to_lds_b128((gvec_p)gp, (lvec_p)lp, 0, 0);
#else
        *(f4*)lp = *(const f4*)gp;
#endif
    }
#ifdef HAVE_ASYNC_COPY
#if __has_builtin(__builtin_amdgcn_s_wait_asynccnt)
    __builtin_amdgcn_s_wait_asynccnt(0);
#else
    asm volatile("s_wait_asynccnt 0" ::: "memory");
#endif
#endif
    __syncthreads();

    const int lane = t & 31;
    const int wave = t >> 5;
    const int n0   = (blockIdx.y * 8 + wave) * 16;
    const int m    = lane & 15;   // A row (lanes 0-15 / 16-31 both map M=0..15)
    const int kh   = lane >> 4;   // K half: 0 -> K=0,1 ; 1 -> K=2,3

    v8f acc = {0.f, 0.f, 0.f, 0.f, 0.f, 0.f, 0.f, 0.f};
    const float* Wp = W + n0 + m;             // column n = lane & 15

    for (int k = 0; k < K; k += 4) {
        v2f a, b;
        a.x = lds_A[m * KP + k + 2 * kh + 0];
        a.y = lds_A[m * KP + k + 2 * kh + 1];
        b.x = Wp[(size_t)(k + 2 * kh + 0) * OUT];
        b.y = Wp[(size_t)(k + 2 * kh + 1) * OUT];
        acc = __builtin_amdgcn_wmma_f32_16x16x4_f32(false, a, false, b,
                                                    (short)0, acc, false, false);
    }

    // D layout: lane n = lane&15 ; VGPR r holds row r + 8*kh
    float* Hp = H + (size_t)m0 * OUT + n0 + m;
#pragma unroll
    for (int r = 0; r < 8; ++r)
        Hp[(size_t)(r + 8 * kh) * OUT] = acc[r];
}

// ---------------- self-loop init: A[i,f] = H[i,f] * dinv[i]^2 ----------------

template <int FEAT>
__global__ void k_init_self(const float* __restrict__ H,
                            const float* __restrict__ dinv, float* A) {
    size_t idx = (size_t)blockIdx.x * 256 + threadIdx.x;
    if (idx >= (size_t)N_NODES * FEAT) return;
    int i = (int)(idx / FEAT);
    float d = dinv[i];
    A[idx] = H[idx] * d * d;
}

// ---------------- edge scatter: one wave per edge, fp32 atomics (L2-resident) ----

template <int FEAT>
__global__ __launch_bounds__(256) void k_scatter(const long long* __restrict__ row,
                                                 const long long* __restrict__ col,
                                                 const float* __restrict__ norm,
                                                 const float* __restrict__ H,
                                                 float* A) {
    int e = blockIdx.x * 8 + (threadIdx.x >> 5);
    if (e >= E_EDGES) return;
    int lane = threadIdx.x & 31;
    int r = (int)row[e], c = (int)col[e];
    float nv = norm[e];
    const float* src = H + (size_t)r * FEAT;
    float* dst = A + (size_t)c * FEAT;
#pragma unroll
    for (int f = lane; f < FEAT; f += 32)
        atomicAdd(&dst[f], src[f] * nv);
}

// ---------------- bias + relu in place ----------------

template <int FEAT>
__global__ void k_bias_relu(float* A, const float* __restrict__ b) {
    size_t idx = (size_t)blockIdx.x * 256 + threadIdx.x;
    if (idx >= (size_t)N_NODES * FEAT) return;
    int f = (int)(idx % FEAT);
    float v = A[idx] + b[f];
    A[idx] = v > 0.f ? v : 0.f;
}

// ---------------- mean over nodes ----------------

__global__ void k_zero_gsum(float* gsum) { gsum[threadIdx.x] = 0.f; }

__global__ void k_gsum(const float* __restrict__ nreps, float* gsum) {
    int f  = threadIdx.x;            // 128 threads: one feature per thread
    int r0 = blockIdx.x * 256;
    int r1 = r0 + 256; if (r1 > N_NODES) r1 = N_NODES;
    float s = 0.f;
    for (int r = r0; r < r1; ++r) s += nreps[(size_t)r * HID2 + f];
    atomicAdd(&gsum[f], s);
}

// ---------------- g_rep, fc, softmax ----------------

__global__ void k_final(const float* __restrict__ gsum,
                        const float* __restrict__ Wfc,
                        const float* __restrict__ bfc, float* out) {
    __shared__ float l0[128], l1[128];
    int f = threadIdx.x;             // 128 threads
    float g = gsum[f] * (1.0f / (float)N_NODES);
    out[2 + (size_t)N_NODES * HID2 + f] = g;           // g_rep
    l0[f] = g * Wfc[f * 2 + 0];
    l1[f] = g * Wfc[f * 2 + 1];
    __syncthreads();
    for (int s = 64; s > 0; s >>= 1) {
        if (f < s) { l0[f] += l0[f + s]; l1[f] += l1[f + s]; }
        __syncthreads();
    }
    if (f == 0) {
        float z0 = l0[0] + bfc[0], z1 = l1[0] + bfc[1];
        float mx = z0 > z1 ? z0 : z1;
        float e0 = __expf(z0 - mx), e1 = __expf(z1 - mx);
        float inv = 1.f / (e0 + e1);
        out[0] = e0 * inv;
        out[1] = e1 * inv;
    }
}

// ---------------- host side ----------------

extern "C" void kernel_launch(void* const* d_in, const int* in_sizes, int n_in,
                              void* d_out, int out_size, void* d_ws, size_t ws_size,
                              hipStream_t stream) {
    const float* x   = (const float*)d_in[0];
    const float* W1  = (const float*)d_in[1];
    const float* b1  = (const float*)d_in[2];
    const float* W2  = (const float*)d_in[3];
    const float* b2  = (const float*)d_in[4];
    const float* Wfc = (const float*)d_in[5];
    const float* bfc = (const float*)d_in[6];
    const long long* ei   = (const long long*)d_in[7];
    const long long* rowp = ei;
    const long long* colp = ei + E_EDGES;
    float* out = (float*)d_out;

    char* ws = (char*)d_ws;
    size_t off = 0;
    auto carve = [&](size_t bytes) -> float* {
        float* p = (float*)(ws + off);
        off = (off + bytes + 255) & ~(size_t)255;
        return p;
    };
    float* deg  = carve((size_t)N_NODES * 4);            // deg -> dinv (in place)
    float* norm = carve((size_t)E_EDGES * 4);
    float* H    = carve((size_t)N_NODES * HID * 4);      // H1, later reused as H2
    float* A1   = carve((size_t)N_NODES * HID * 4);
    float* gsum = carve(128 * 4);
    (void)ws_size; (void)in_sizes; (void)n_in; (void)out_size;

    float* nreps = out + 2;                              // n_reps region of d_out

    const int TB = 256;
    k_fill1 <<<(N_NODES + TB - 1) / TB, TB, 0, stream>>>(deg);
    k_degree<<<(E_EDGES + TB - 1) / TB, TB, 0, stream>>>(colp, deg);
    k_dinv  <<<(N_NODES + TB - 1) / TB, TB, 0, stream>>>(deg);
    k_norm  <<<(E_EDGES + TB - 1) / TB, TB, 0, stream>>>(rowp, colp, deg, norm);

    // layer 1: H1 = X @ W1 ; A1 = scatter(norm * H1) ; relu(A1 + b1)
    k_gemm_wmma<IN_DIM, HID><<<dim3(N_NODES / 16, HID / 128), TB, 0, stream>>>(x, W1, H);
    k_init_self<HID><<<(N_NODES * HID + TB - 1) / TB, TB, 0, stream>>>(H, deg, A1);
    k_scatter<HID><<<(E_EDGES + 7) / 8, TB, 0, stream>>>(rowp, colp, norm, H, A1);
    k_bias_relu<HID><<<(N_NODES * HID + TB - 1) / TB, TB, 0, stream>>>(A1, b1);

    // layer 2: H2 = A1 @ W2 (reuse H buffer) ; n_reps = relu(scatter + b2)
    k_gemm_wmma<HID, HID2><<<dim3(N_NODES / 16, HID2 / 128), TB, 0, stream>>>(A1, W2, H);
    k_init_self<HID2><<<(N_NODES * HID2 + TB - 1) / TB, TB, 0, stream>>>(H, deg, nreps);
    k_scatter<HID2><<<(E_EDGES + 7) / 8, TB, 0, stream>>>(rowp, colp, norm, H, nreps);
    k_bias_relu<HID2><<<(N_NODES * HID2 + TB - 1) / TB, TB, 0, stream>>>(nreps, b2);

    // readout: g_rep = mean(n_reps), logits = g_rep @ Wfc + bfc, softmax
    k_zero_gsum<<<1, 128, 0, stream>>>(gsum);
    k_gsum<<<(N_NODES + 255) / 256, 128, 0, stream>>>(nreps, gsum);
    k_final<<<1, 128, 0, stream>>>(gsum, Wfc, bfc, out);
}